// CBOWCellEncoder_31430570672172
// MI455X (gfx1250) — compile-verified
//
#include <hip/hip_runtime.h>
#include <math.h>

#define GENES   4096
#define BATCH   256
#define DGENE   256

typedef __attribute__((ext_vector_type(2))) float v2f;
typedef __attribute__((ext_vector_type(8))) float v8f;

// ---------------------------------------------------------------------------
// Kernel 1: weighted embedding pool + log-libsize.
// One block per batch row. Pass 1: cache log1p(x) in LDS, reduce sums.
// Pass 2: gather emb rows with float4 (B128) loads; 64 lanes x float4 = 1KB row,
// 4 rows in flight per iteration (r = tid/64), x4 unroll -> 4 gathers in flight
// per wave. emb_table (30.7MB) is L2-resident (192MB L2), so this phase is
// L2-gather-bandwidth bound, not HBM bound.
// ---------------------------------------------------------------------------
__global__ __launch_bounds__(256) void pool_kernel(
    const float* __restrict__ x, const int* __restrict__ gidx,
    const float* __restrict__ emb, float* __restrict__ h_pool,
    float* __restrict__ log_lib)
{
    __shared__ float lw[GENES];      // 16 KB: log1p weights
    __shared__ float rsum[256];
    __shared__ float rsx[256];
    __shared__ float4 part[256];     // 4 KB: cross-r partials

    const int b = blockIdx.x;
    const int t = threadIdx.x;

    // Pass 1: log1p weights + row sums
    float s_lw = 0.0f, s_x = 0.0f;
    for (int g = t; g < GENES; g += 256) {
        float xv = x[(size_t)b * GENES + g];
        float l  = log1pf(xv);
        lw[g] = l;
        s_lw += l;
        s_x  += xv;
    }
    rsum[t] = s_lw; rsx[t] = s_x;
    __syncthreads();
    for (int off = 128; off > 0; off >>= 1) {
        if (t < off) { rsum[t] += rsum[t + off]; rsx[t] += rsx[t + off]; }
        __syncthreads();
    }
    const float inv = 1.0f / (rsum[0] + 1e-8f);
    if (t == 0) log_lib[b] = log1pf(rsx[0]);

    // Pass 2: gather-accumulate. thread = (r, c): r=row-in-flight, c=dim/4
    const int r = t >> 6;       // 0..3
    const int c = t & 63;       // 0..63  -> dims 4c..4c+3
    const int* __restrict__ gb = gidx + (size_t)b * GENES;
    const float* __restrict__ ec = emb + (size_t)c * 4;
    float a0 = 0.f, a1 = 0.f, a2 = 0.f, a3 = 0.f;
#pragma unroll 4
    for (int g = r; g < GENES; g += 4) {
        const int   idx = gb[g];            // broadcast within r-group
        const float wv  = lw[g] * inv;
        const float4 e  = *(const float4*)(ec + (size_t)idx * DGENE);
        a0 = fmaf(wv, e.x, a0);
        a1 = fmaf(wv, e.y, a1);
        a2 = fmaf(wv, e.z, a2);
        a3 = fmaf(wv, e.w, a3);
    }
    part[t] = make_float4(a0, a1, a2, a3);
    __syncthreads();
    if (r == 0) {
        float4 p0 = part[c], p1 = part[64 + c], p2 = part[128 + c], p3 = part[192 + c];
        float4 o = make_float4(p0.x + p1.x + p2.x + p3.x,
                               p0.y + p1.y + p2.y + p3.y,
                               p0.z + p1.z + p2.z + p3.z,
                               p0.w + p1.w + p2.w + p3.w);
        *(float4*)(h_pool + (size_t)b * DGENE + c * 4) = o;
    }
}

// ---------------------------------------------------------------------------
// Kernel 2: fp32 WMMA GEMM:  C = act(A[MxK] @ W[KxN] + bias [+ u ⊗ vrow])
// K, N compile-time so all inner-loop loads use immediate offsets from two
// loop-invariant base pointers (no per-iteration 64-bit address math).
// One wave per 16x16 output tile; K-loop of V_WMMA_F32_16X16X4_F32.
// A-frag: lane (lg=lane>>4, rin=lane&15) holds A[row=rin][k=2*lg+{0,1}] (float2).
// B-frag: holds W[k=2*lg+{0,1}][n=rin] (two coalesced b32 loads).
// C/D:    acc[v] = C[m = v + 8*lg][n = rin].
// ---------------------------------------------------------------------------
template <int K, int N, bool RELU, bool RANK1>
__global__ __launch_bounds__(128) void gemm_wmma_f32(
    const float* __restrict__ A, const float* __restrict__ W,
    const float* __restrict__ bias, const float* __restrict__ u,
    const float* __restrict__ vrow, float* __restrict__ C)
{
    const int wave = threadIdx.x >> 5;
    const int lane = threadIdx.x & 31;
    const int tile = blockIdx.x * 4 + wave;
    constexpr int tilesN = N >> 4;
    const int tm = tile / tilesN;
    const int tn = tile % tilesN;

    const int lg  = lane >> 4;           // 0 or 1
    const int rin = lane & 15;           // row (A) / col (B)

    v8f acc = {};
    const float* __restrict__ Ap = A + (size_t)(tm * 16 + rin) * K + 2 * lg;
    const float* __restrict__ Wp = W + (size_t)(2 * lg) * N + tn * 16 + rin;

#pragma unroll 8
    for (int kk = 0; kk < K; kk += 4) {
        v2f a;
        a.x = Ap[kk];
        a.y = Ap[kk + 1];
        v2f bf;
        bf.x = Wp[kk * N];
        bf.y = Wp[(kk + 1) * N];
        acc = __builtin_amdgcn_wmma_f32_16x16x4_f32(
            /*neg_a=*/false, a, /*neg_b=*/false, bf,
            /*c_mod=*/(short)0, acc, /*reuse_a=*/false, /*reuse_b=*/false);
    }

    const int colG = tn * 16 + rin;
    const float bia = bias[colG];
    const float vr  = RANK1 ? vrow[colG] : 0.0f;
#pragma unroll
    for (int v = 0; v < 8; ++v) {
        const int m = tm * 16 + v + 8 * lg;
        float val = acc[v] + bia;
        if (RANK1) val = fmaf(u[m], vr, val);
        if (RELU)  val = fmaxf(val, 0.0f);
        C[(size_t)m * N + colG] = val;
    }
}

// ---------------------------------------------------------------------------
extern "C" void kernel_launch(void* const* d_in, const int* in_sizes, int n_in,
                              void* d_out, int out_size, void* d_ws, size_t ws_size,
                              hipStream_t stream) {
    const float* x    = (const float*)d_in[0];
    const int*   gidx = (const int*)  d_in[1];
    const float* emb  = (const float*)d_in[2];
    const float* W1   = (const float*)d_in[3];   // (257, 256) row-major
    const float* b1   = (const float*)d_in[4];
    const float* W2   = (const float*)d_in[5];   // (256, 256)
    const float* b2   = (const float*)d_in[6];
    const float* Wmu  = (const float*)d_in[7];   // (256, 64)
    const float* bmu  = (const float*)d_in[8];
    const float* Wlv  = (const float*)d_in[9];   // (256, 64)
    const float* blv  = (const float*)d_in[10];

    float* out    = (float*)d_out;
    float* mu     = out;                    // 256*64
    float* lv     = out + 256 * 64;         // 256*64
    float* h_pool = out + 2 * 256 * 64;     // 256*256 (output #3)

    float* ws      = (float*)d_ws;
    float* log_lib = ws;                    // 256
    float* H1      = ws + 256;              // 256*256
    float* H2      = H1 + 256 * 256;        // 256*256

    // Phase 1: gather-pool (L2-resident emb_table; ~1 GiB gathered reads)
    pool_kernel<<<BATCH, 256, 0, stream>>>(x, gidx, emb, h_pool, log_lib);

    // Phase 2: WMMA MLP head
    // H1 = relu(h_pool @ W1[0:256] + log_lib ⊗ W1[256,:] + b1)
    gemm_wmma_f32<256, 256, true, true><<<64, 128, 0, stream>>>(
        h_pool, W1, b1, log_lib, W1 + 256 * 256, H1);
    // H2 = relu(H1 @ W2 + b2)
    gemm_wmma_f32<256, 256, true, false><<<64, 128, 0, stream>>>(
        H1, W2, b2, nullptr, nullptr, H2);
    // mu = H2 @ Wmu + bmu ; lv = H2 @ Wlv + blv
    gemm_wmma_f32<256, 64, false, false><<<16, 128, 0, stream>>>(
        H2, Wmu, bmu, nullptr, nullptr, mu);
    gemm_wmma_f32<256, 64, false, false><<<16, 128, 0, stream>>>(
        H2, Wlv, blv, nullptr, nullptr, lv);
}